// _sample_and_group_serial_58514634441108
// MI455X (gfx1250) — compile-verified
//
#include <hip/hip_runtime.h>
#include <hip/hip_bf16.h>
#include <math.h>
#include <stdint.h>

// ---------------------------------------------------------------------------
// Problem constants (match reference setup_inputs)
// ---------------------------------------------------------------------------
static constexpr int B_  = 4;
static constexpr int N_  = 8192;   // source points
static constexpr int M_  = 2048;   // NPOINT (sampled)
static constexpr int KNN = 16;     // N_KNN
static constexpr int GK  = 8;      // END_K groups (pairs of sorted neighbors)
static constexpr int DIM = 64;     // feature dim
static constexpr int HID = 128;    // 2*DIM attention hidden
static constexpr float EPSV = 1e-5f;

typedef __attribute__((ext_vector_type(2))) float v2f;
typedef __attribute__((ext_vector_type(8))) float v8f;

// Async global->LDS copy of one B32 per active lane (gfx1250 VGLOBAL async op,
// tracked by ASYNCcnt).  lds_off = low 32 bits of the generic shared pointer
// (shared->generic cast puts the LDS byte offset in the low half).
__device__ __forceinline__ void async_load_b32(const float* gptr, const float* lptr)
{
    const unsigned lds_off = (unsigned)(uintptr_t)lptr;
    asm volatile("global_load_async_to_lds_b32 %0, %1, off"
                 :: "v"(lds_off), "v"(gptr) : "memory");
}
__device__ __forceinline__ void wait_async0()
{
    asm volatile("s_wait_asynccnt 0x0" ::: "memory");
}

// ---------------------------------------------------------------------------
// 1) Farthest point sampling: one 1024-thread block per batch.
//    Distances live in registers (8 points / thread), argmax via LDS tree
//    reduction, selection recorded in an LDS bitmap, then a block prefix
//    scan over the bitmap emits the indices ALREADY SORTED (matches
//    jnp.sort(fps(...)) since FPS indices are distinct).
// ---------------------------------------------------------------------------
__global__ void fps_kernel(const float* __restrict__ xyz, int* __restrict__ sidx)
{
    const int b = blockIdx.x;
    const int t = threadIdx.x;                 // 0..1023
    const float* xb = xyz + (size_t)b * 3 * N_;

    __shared__ float s_val[1024];
    __shared__ int   s_idx[1024];
    __shared__ unsigned s_flag[N_ / 32];       // 8192-bit selection bitmap
    __shared__ int   s_last;

    for (int i = t; i < N_ / 32; i += 1024) s_flag[i] = 0u;

    float px[8], py[8], pz[8], dd[8];
#pragma unroll
    for (int j = 0; j < 8; ++j) {
        const int n = t * 8 + j;
        px[j] = xb[n]; py[j] = xb[N_ + n]; pz[j] = xb[2 * N_ + n];
        dd[j] = 1e10f;
    }
    if (t == 0) { s_flag[0] = 1u; s_last = 0; }   // first index is 0 (like ref)
    __syncthreads();

    int last = 0;
    for (int it = 1; it < M_; ++it) {
        const float lx = xb[last], ly = xb[N_ + last], lz = xb[2 * N_ + last];
        float bv = -1.0f; int bn = t * 8;
#pragma unroll
        for (int j = 0; j < 8; ++j) {
            const float dx = px[j] - lx, dy = py[j] - ly, dz = pz[j] - lz;
            const float dn = dx * dx + dy * dy + dz * dz;
            dd[j] = fminf(dd[j], dn);
            if (dd[j] > bv) { bv = dd[j]; bn = t * 8 + j; }
        }
        s_val[t] = bv; s_idx[t] = bn;
        __syncthreads();
        for (int off = 512; off > 0; off >>= 1) {
            if (t < off) {
                if (s_val[t + off] > s_val[t]) {
                    s_val[t] = s_val[t + off]; s_idx[t] = s_idx[t + off];
                }
            }
            __syncthreads();
        }
        if (t == 0) {
            const int sel = s_idx[0];
            s_last = sel;
            s_flag[sel >> 5] |= (1u << (sel & 31));
        }
        __syncthreads();
        last = s_last;
    }

    // compact bitmap -> sorted indices. Thread t owns contiguous points
    // [t*8, t*8+8) == one byte of the bitmap.
    const unsigned w = (s_flag[t >> 2] >> ((t & 3) * 8)) & 0xFFu;
    const int cnt = __popc(w);
    s_idx[t] = cnt;
    __syncthreads();
    for (int off = 1; off < 1024; off <<= 1) {
        const int v = (t >= off) ? s_idx[t - off] : 0;
        __syncthreads();
        s_idx[t] += v;
        __syncthreads();
    }
    int base = s_idx[t] - cnt;                 // exclusive prefix
    int* sb = sidx + (size_t)b * M_;
#pragma unroll
    for (int j = 0; j < 8; ++j)
        if (w & (1u << j)) sb[base++] = t * 8 + j;
}

// ---------------------------------------------------------------------------
// 2) Gather new_xyz (-> d_out) and new_feat (-> ws)
// ---------------------------------------------------------------------------
__global__ void gather_kernel(const float* __restrict__ xyz,
                              const float* __restrict__ points,
                              const int* __restrict__ sidx,
                              float* __restrict__ out_xyz,
                              float* __restrict__ new_feat)
{
    const int gid = blockIdx.x * blockDim.x + threadIdx.x;
    if (gid >= B_ * M_) return;
    const int b = gid / M_, m = gid % M_;
    const int s = sidx[(size_t)b * M_ + m];
#pragma unroll
    for (int c = 0; c < 3; ++c)
        out_xyz[(size_t)b * 3 * M_ + c * M_ + m] = xyz[(size_t)b * 3 * N_ + c * N_ + s];
    const float* pb = points + (size_t)b * DIM * N_;
    float* nf = new_feat + (size_t)b * DIM * M_;
    for (int c = 0; c < DIM; ++c) nf[(size_t)c * M_ + m] = pb[(size_t)c * N_ + s];
}

// ---------------------------------------------------------------------------
// 3) 1x1 conv as GEMM with V_WMMA_F32_16X16X4_F32.
//    Y[b, o, p] = sum_i W[o,i] * X[b,i,p] + bias[o]
//    One wave (32 lanes) computes a 16(O)x16(P) tile; K stepped by 4.
//    A layout (16x4 f32, ISA 7.12.2): lanes 0-15 hold K=0/1, lanes 16-31 K=2/3.
// ---------------------------------------------------------------------------
template <int I>
__global__ void conv1x1_wmma_kernel(const float* __restrict__ W,
                                    const float* __restrict__ bias,
                                    const float* __restrict__ X,
                                    float* __restrict__ Y,
                                    int O, int P)
{
    const int lane = threadIdx.x;              // 0..31, wave32
    const int p0 = blockIdx.x * 16;
    const int o0 = blockIdx.y * 16;
    const int b  = blockIdx.z;
    const float* Xb = X + (size_t)b * I * P;
    float* Yb = Y + (size_t)b * O * P;
    const int half = lane >> 4;                // 0: K lo / rows 0-7, 1: K hi / rows 8-15
    const int l16  = lane & 15;

    v8f acc;
#pragma unroll
    for (int i = 0; i < 8; ++i) acc[i] = bias[o0 + (half ? i + 8 : i)];

#pragma unroll
    for (int k0 = 0; k0 < I; k0 += 4) {
        v2f a, bf;
        const int arow = o0 + l16;
        a.x  = W[(size_t)arow * I + k0 + 2 * half + 0];
        a.y  = W[(size_t)arow * I + k0 + 2 * half + 1];
        bf.x = Xb[(size_t)(k0 + 2 * half + 0) * P + p0 + l16];
        bf.y = Xb[(size_t)(k0 + 2 * half + 1) * P + p0 + l16];
        acc = __builtin_amdgcn_wmma_f32_16x16x4_f32(false, a, false, bf,
                                                    (short)0, acc, false, false);
    }
#pragma unroll
    for (int i = 0; i < 8; ++i) {
        const int r = o0 + (half ? i + 8 : i);
        Yb[(size_t)r * P + p0 + l16] = acc[i];
    }
}

// ---------------------------------------------------------------------------
// 4) 16-NN (sorted indices). One thread per query point; source xyz staged
//    through LDS in 256-point chunks via ASYNC global->LDS copies;
//    registers hold the running top-16.
// ---------------------------------------------------------------------------
__global__ void knn_kernel(const float* __restrict__ xyz,
                           const float* __restrict__ new_xyz,
                           int* __restrict__ knn)
{
    const int tid = threadIdx.x;
    const int m = blockIdx.x * 256 + tid;
    const int b = blockIdx.y;
    const float* xb = xyz + (size_t)b * 3 * N_;
    const float qx = new_xyz[(size_t)b * 3 * M_ + m];
    const float qy = new_xyz[(size_t)b * 3 * M_ + M_ + m];
    const float qz = new_xyz[(size_t)b * 3 * M_ + 2 * M_ + m];

    __shared__ float sx[256], sy[256], sz[256];
    float bd[KNN]; int bi[KNN];
#pragma unroll
    for (int q = 0; q < KNN; ++q) { bd[q] = 3.0e38f; bi[q] = 0; }

    for (int base = 0; base < N_; base += 256) {
        __syncthreads();
        // async global->LDS staging of this xyz chunk (ASYNCcnt-tracked)
        async_load_b32(xb + base + tid,          &sx[tid]);
        async_load_b32(xb + N_ + base + tid,     &sy[tid]);
        async_load_b32(xb + 2 * N_ + base + tid, &sz[tid]);
        if (base + 256 < N_) __builtin_prefetch(xb + base + 256 + tid, 0, 0);
        wait_async0();
        __syncthreads();
        for (int j = 0; j < 256; ++j) {
            const float dx = qx - sx[j], dy = qy - sy[j], dz = qz - sz[j];
            const float d = dx * dx + dy * dy + dz * dz;
            if (d < bd[KNN - 1]) {
                float cd = d; int ci = base + j;
#pragma unroll
                for (int q = 0; q < KNN; ++q) {
                    if (cd < bd[q]) {
                        const float td = bd[q]; const int ti = bi[q];
                        bd[q] = cd; bi[q] = ci; cd = td; ci = ti;
                    }
                }
            }
        }
    }
    // sort the 16 nearest indices ascending (matches jnp.sort(idx, -1))
#pragma unroll
    for (int pass = 0; pass < KNN; ++pass)
#pragma unroll
        for (int q = 0; q < KNN - 1; ++q)
            if (bi[q] > bi[q + 1]) { const int ti = bi[q]; bi[q] = bi[q + 1]; bi[q + 1] = ti; }

    int* kb = knn + ((size_t)b * M_ + m) * KNN;
#pragma unroll
    for (int q = 0; q < KNN; ++q) kb[q] = bi[q];
}

// ---------------------------------------------------------------------------
// 5) Fused positional MLP + key gather + group-max.
//    One 64-thread block per (m,b): builds rel(3x16) and the ReLU(BN(3->32))
//    hidden in LDS, then thread d computes pos(d,k), qk/val group maxima.
// ---------------------------------------------------------------------------
__global__ void posqk_kernel(const float* __restrict__ xyz,
                             const float* __restrict__ new_xyz,
                             const int* __restrict__ knn,
                             const float* __restrict__ p1w, const float* __restrict__ p1b,
                             const float* __restrict__ pbg, const float* __restrict__ pbb,
                             const float* __restrict__ pbm, const float* __restrict__ pbv,
                             const float* __restrict__ p2w, const float* __restrict__ p2b,
                             const float* __restrict__ key_f,
                             const float* __restrict__ query,
                             const float* __restrict__ value,
                             float* __restrict__ qk, float* __restrict__ val)
{
    const int m = blockIdx.x, b = blockIdx.y, tid = threadIdx.x;  // 64 threads
    __shared__ int   s_idx[KNN];
    __shared__ float s_rel[3][KNN];
    __shared__ float s_h[32][KNN];

    if (tid < KNN) s_idx[tid] = knn[((size_t)b * M_ + m) * KNN + tid];
    __syncthreads();
    if (tid < 3 * KNN) {
        const int c = tid >> 4, k = tid & 15;
        s_rel[c][k] = new_xyz[(size_t)b * 3 * M_ + c * M_ + m]
                    - xyz[(size_t)b * 3 * N_ + c * N_ + s_idx[k]];
    }
    __syncthreads();
    for (int e = tid; e < 32 * KNN; e += 64) {
        const int hc = e >> 4, k = e & 15;
        float v = p1b[hc]
                + p1w[hc * 3 + 0] * s_rel[0][k]
                + p1w[hc * 3 + 1] * s_rel[1][k]
                + p1w[hc * 3 + 2] * s_rel[2][k];
        const float inv = pbg[hc] * rsqrtf(pbv[hc] + EPSV);
        v = v * inv + (pbb[hc] - pbm[hc] * inv);
        s_h[hc][k] = fmaxf(v, 0.0f);
    }
    __syncthreads();

    const int d = tid;
    const float qd = query[((size_t)b * DIM + d) * M_ + m];
    const float vd = value[((size_t)b * DIM + d) * M_ + m];
    const float* kf = key_f + ((size_t)b * DIM + d) * N_;
    const float* w2 = p2w + d * 32;
    const float b2 = p2b[d];
    float* qkb = qk  + (((size_t)b * DIM + d) * M_ + m) * GK;
    float* vlb = val + (((size_t)b * DIM + d) * M_ + m) * GK;

#pragma unroll
    for (int g = 0; g < GK; ++g) {
        float qm = -3.0e38f, vm = -3.0e38f;
#pragma unroll
        for (int r = 0; r < 2; ++r) {
            const int k = g * 2 + r;
            float pos = b2;
#pragma unroll
            for (int hc = 0; hc < 32; ++hc) pos += w2[hc] * s_h[hc][k];
            const float kg = kf[s_idx[k]];
            qm = fmaxf(qm, qd - kg + pos);
            vm = fmaxf(vm, vd - kg + pos);
        }
        qkb[g] = qm; vlb[g] = vm;
    }
}

// ---------------------------------------------------------------------------
// 6) Fused attention MLP (64->128 BN/ReLU ->64) over pixels P = M*GK.
//    Block = 128 threads = 4 waves; 16-pixel tile; hidden tile lives in LDS.
//    Activation tile staged into LDS with ASYNC global->LDS copies.
//    GEMM1: each wave does 2 of the 8 (16x16) output tiles; GEMM2: 1 tile/wave.
// ---------------------------------------------------------------------------
__global__ void attn_mlp_kernel(const float* __restrict__ qk,
                                const float* __restrict__ a1w, const float* __restrict__ a1b,
                                const float* __restrict__ abg, const float* __restrict__ abb,
                                const float* __restrict__ abm, const float* __restrict__ abv,
                                const float* __restrict__ a2w, const float* __restrict__ a2b,
                                float* __restrict__ attn)
{
    const int P = M_ * GK;
    const int p0 = blockIdx.x * 16;
    const int b  = blockIdx.y;
    const int tid = threadIdx.x;
    const int lane = tid & 31, wave = tid >> 5;
    const int half = lane >> 4, l16 = lane & 15;

    __shared__ float s_x[DIM][16];
    __shared__ float s_h[HID][16];

    const float* Xb = qk + (size_t)b * DIM * P;
#pragma unroll
    for (int e0 = 0; e0 < DIM * 16; e0 += 128) {
        const int e = e0 + tid;
        const int c = e >> 4, p = e & 15;
        async_load_b32(&Xb[(size_t)c * P + p0 + p], &s_x[c][p]);
    }
    wait_async0();
    __syncthreads();

    // GEMM1: H = ReLU(BN(a1w @ X))  (128x16)
#pragma unroll
    for (int t = 0; t < 2; ++t) {
        const int o0 = (wave + 4 * t) * 16;
        v8f acc;
#pragma unroll
        for (int i = 0; i < 8; ++i) acc[i] = a1b[o0 + (half ? i + 8 : i)];
#pragma unroll
        for (int k0 = 0; k0 < DIM; k0 += 4) {
            v2f a, bf;
            a.x  = a1w[(size_t)(o0 + l16) * DIM + k0 + 2 * half + 0];
            a.y  = a1w[(size_t)(o0 + l16) * DIM + k0 + 2 * half + 1];
            bf.x = s_x[k0 + 2 * half + 0][l16];
            bf.y = s_x[k0 + 2 * half + 1][l16];
            acc = __builtin_amdgcn_wmma_f32_16x16x4_f32(false, a, false, bf,
                                                        (short)0, acc, false, false);
        }
#pragma unroll
        for (int i = 0; i < 8; ++i) {
            const int r = o0 + (half ? i + 8 : i);
            const float inv = abg[r] * rsqrtf(abv[r] + EPSV);
            const float v = acc[i] * inv + (abb[r] - abm[r] * inv);
            s_h[r][l16] = fmaxf(v, 0.0f);
        }
    }
    __syncthreads();

    // GEMM2: A = a2w @ H  (64x16)
    {
        const int o0 = wave * 16;
        v8f acc;
#pragma unroll
        for (int i = 0; i < 8; ++i) acc[i] = a2b[o0 + (half ? i + 8 : i)];
#pragma unroll
        for (int k0 = 0; k0 < HID; k0 += 4) {
            v2f a, bf;
            a.x  = a2w[(size_t)(o0 + l16) * HID + k0 + 2 * half + 0];
            a.y  = a2w[(size_t)(o0 + l16) * HID + k0 + 2 * half + 1];
            bf.x = s_h[k0 + 2 * half + 0][l16];
            bf.y = s_h[k0 + 2 * half + 1][l16];
            acc = __builtin_amdgcn_wmma_f32_16x16x4_f32(false, a, false, bf,
                                                        (short)0, acc, false, false);
        }
        float* Ab = attn + (size_t)b * DIM * P;
#pragma unroll
        for (int i = 0; i < 8; ++i) {
            const int r = o0 + (half ? i + 8 : i);
            Ab[(size_t)r * P + p0 + l16] = acc[i];
        }
    }
}

// ---------------------------------------------------------------------------
// 7) Softmax over GK, aggregate with val, residual ew-conv, final uw-conv.
//    128-thread block per (m,b); matvecs staged through LDS.
// ---------------------------------------------------------------------------
__global__ void final_kernel(const float* __restrict__ attn,
                             const float* __restrict__ val,
                             const float* __restrict__ new_feat,
                             const float* __restrict__ ew, const float* __restrict__ eb,
                             const float* __restrict__ uw, const float* __restrict__ ub,
                             float* __restrict__ out_points)
{
    const int m = blockIdx.x, b = blockIdx.y, tid = threadIdx.x;  // 128 threads
    __shared__ float s_agg[DIM];
    __shared__ float s_y[DIM];

    if (tid < DIM) {
        const int d = tid;
        const float* lg = attn + (((size_t)b * DIM + d) * M_ + m) * GK;
        const float* vv = val  + (((size_t)b * DIM + d) * M_ + m) * GK;
        float mx = lg[0];
#pragma unroll
        for (int g = 1; g < GK; ++g) mx = fmaxf(mx, lg[g]);
        float e[GK]; float s = 0.0f;
#pragma unroll
        for (int g = 0; g < GK; ++g) { e[g] = __expf(lg[g] - mx); s += e[g]; }
        const float invs = 1.0f / s;
        float agg = 0.0f;
#pragma unroll
        for (int g = 0; g < GK; ++g) agg += e[g] * invs * vv[g];
        s_agg[d] = agg;
    }
    __syncthreads();
    if (tid < DIM) {
        const int d = tid;
        float y = eb[d] + new_feat[((size_t)b * DIM + d) * M_ + m];
#pragma unroll
        for (int i = 0; i < DIM; ++i) y += ew[d * DIM + i] * s_agg[i];
        s_y[d] = y;
    }
    __syncthreads();
    {
        const int o = tid;   // 0..127
        float v = ub[o];
#pragma unroll
        for (int i = 0; i < DIM; ++i) v += uw[o * DIM + i] * s_y[i];
        out_points[((size_t)b * 128 + o) * M_ + m] = v;
    }
}

// ---------------------------------------------------------------------------
// Launch
// ---------------------------------------------------------------------------
extern "C" void kernel_launch(void* const* d_in, const int* in_sizes, int n_in,
                              void* d_out, int out_size, void* d_ws, size_t ws_size,
                              hipStream_t stream)
{
    const float* xyz    = (const float*)d_in[0];
    const float* points = (const float*)d_in[1];
    const float* vw  = (const float*)d_in[2];  const float* vb  = (const float*)d_in[3];
    const float* kw  = (const float*)d_in[4];  const float* kb  = (const float*)d_in[5];
    const float* qw  = (const float*)d_in[6];  const float* qb  = (const float*)d_in[7];
    const float* p1w = (const float*)d_in[8];  const float* p1b = (const float*)d_in[9];
    const float* pbg = (const float*)d_in[10]; const float* pbb = (const float*)d_in[11];
    const float* pbm = (const float*)d_in[12]; const float* pbv = (const float*)d_in[13];
    const float* p2w = (const float*)d_in[14]; const float* p2b = (const float*)d_in[15];
    const float* a1w = (const float*)d_in[16]; const float* a1b = (const float*)d_in[17];
    const float* abg = (const float*)d_in[18]; const float* abb = (const float*)d_in[19];
    const float* abm = (const float*)d_in[20]; const float* abv = (const float*)d_in[21];
    const float* a2w = (const float*)d_in[22]; const float* a2b = (const float*)d_in[23];
    const float* ew  = (const float*)d_in[24]; const float* eb  = (const float*)d_in[25];
    const float* uw  = (const float*)d_in[26]; const float* ub  = (const float*)d_in[27];

    float* out_xyz    = (float*)d_out;                       // (B,3,M)
    float* out_points = out_xyz + (size_t)B_ * 3 * M_;       // (B,128,M)

    // workspace layout
    int*   sidx = (int*)d_ws;                                // B*M
    int*   knn  = sidx + (size_t)B_ * M_;                    // B*M*16
    float* new_feat = (float*)(knn + (size_t)B_ * M_ * KNN); // B*64*M
    float* key_f = new_feat + (size_t)B_ * DIM * M_;         // B*64*N
    float* value = key_f + (size_t)B_ * DIM * N_;            // B*64*M
    float* query = value + (size_t)B_ * DIM * M_;            // B*64*M
    float* qk    = query + (size_t)B_ * DIM * M_;            // B*64*M*8
    float* val   = qk    + (size_t)B_ * DIM * M_ * GK;       // B*64*M*8
    float* attn  = val   + (size_t)B_ * DIM * M_ * GK;       // B*64*M*8

    fps_kernel<<<dim3(B_), dim3(1024), 0, stream>>>(xyz, sidx);
    gather_kernel<<<dim3((B_ * M_) / 256), dim3(256), 0, stream>>>(
        xyz, points, sidx, out_xyz, new_feat);

    conv1x1_wmma_kernel<DIM><<<dim3(N_ / 16, DIM / 16, B_), dim3(32), 0, stream>>>(
        kw, kb, points, key_f, DIM, N_);
    conv1x1_wmma_kernel<DIM><<<dim3(M_ / 16, DIM / 16, B_), dim3(32), 0, stream>>>(
        vw, vb, new_feat, value, DIM, M_);
    conv1x1_wmma_kernel<DIM><<<dim3(M_ / 16, DIM / 16, B_), dim3(32), 0, stream>>>(
        qw, qb, new_feat, query, DIM, M_);

    knn_kernel<<<dim3(M_ / 256, B_), dim3(256), 0, stream>>>(xyz, out_xyz, knn);

    posqk_kernel<<<dim3(M_, B_), dim3(64), 0, stream>>>(
        xyz, out_xyz, knn, p1w, p1b, pbg, pbb, pbm, pbv, p2w, p2b,
        key_f, query, value, qk, val);

    attn_mlp_kernel<<<dim3((M_ * GK) / 16, B_), dim3(128), 0, stream>>>(
        qk, a1w, a1b, abg, abb, abm, abv, a2w, a2b, attn);

    final_kernel<<<dim3(M_, B_), dim3(128), 0, stream>>>(
        attn, val, new_feat, ew, eb, uw, ub, out_points);
}